// DWTChannelExtractor_34041910788292
// MI455X (gfx1250) — compile-verified
//
#include <hip/hip_runtime.h>

// 2D Haar DWT, (32,3,512,512) f32 -> (32,12,256,256) f32.
//
// One wave32 owns one full output row (256 pixels = 16 tiles of 16 pixels).
// Phase 1: stage all 16 B-tiles with a clause of 16 global_load_b64 (NT) —
//          4 KB in flight per wave (lanes 0-15 = even input row strip,
//          lanes 16-31 = odd input row strip).
// Phase 2: 16x V_WMMA_F32_16X16X4_F32, each draining loadcnt incrementally:
//   A (16x4) = Haar analysis matrix in rows 0..3 (lane-local, built once)
//   D (16x16) = VGPRs 0..3, lanes 0..15 hold LL/LH/HL/HH of 16 pixels,
//               stored as a 4-store NT clause per tile.
// Streaming kernel: ~192 MB traffic -> ~8.3 us floor at 23.3 TB/s.

typedef __attribute__((ext_vector_type(2))) float v2f;
typedef __attribute__((ext_vector_type(8))) float v8f;

#define IN_W   512
#define IN_H   512
#define OUT_W  256
#define OUT_H  256
#define N_IMG  96                 // 32 batches * 3 channels
#define PLANE  (OUT_H * OUT_W)    // 65536

__global__ __launch_bounds__(256) void
haar_dwt2_wmma_kernel(const float* __restrict__ x, float* __restrict__ out)
{
    const int lane = threadIdx.x & 31;
    const int wid  = (blockIdx.x * 256 + threadIdx.x) >> 5;  // global wave id
    const int i    = wid & (OUT_H - 1);                      // output row
    const int img  = wid >> 8;                               // b*3 + c (0..95)

    const int n  = lane & 15;   // A row / B,D column
    const int hi = lane >> 4;   // 0: even input row, K={0,1}; 1: odd row, K={2,3}

    // ---- A matrix (built once): rows 0..3 = Haar (+-0.5), rows 4..15 = 0 ---
    // sign(m,k) = (-1)^((m&1)*(k>>1) + ((m>>1)&1)*(k&1))
    v2f amat;
    {
        const int k0 = 2 * hi;
        const int k1 = k0 + 1;
        const int e0 = ((n & 1) * (k0 >> 1) + ((n >> 1) & 1) * (k0 & 1)) & 1;
        const int e1 = ((n & 1) * (k1 >> 1) + ((n >> 1) & 1) * (k1 & 1)) & 1;
        amat.x = (n < 4) ? (e0 ? -0.5f : 0.5f) : 0.0f;
        amat.y = (n < 4) ? (e1 ? -0.5f : 0.5f) : 0.0f;
    }

    // ---- base pointers (once per wave) -------------------------------------
    // Input: lane n (lo half) -> {x00,x01} at (2i, 2n); hi half -> (2i+1, 2n)
    const float* src = x + (size_t)img * (IN_H * IN_W)
                         + (size_t)(2 * i + hi) * IN_W
                         + 2 * n;
    // Output: channel = img*4 + r, plane stride 65536 (lanes 0..15 store)
    float* dstBase = out + (size_t)img * 4 * PLANE
                         + (size_t)i * OUT_W + n;

    // ---- Phase 1: stage all 16 B-tiles (one load clause, 16 outstanding) ---
    v2f b[16];
    #pragma unroll
    for (int jt = 0; jt < 16; ++jt)
        b[jt] = __builtin_nontemporal_load((const v2f*)(src + 32 * jt));

    // ---- Phase 2: WMMA + store per tile, draining loadcnt incrementally ----
    #pragma unroll
    for (int jt = 0; jt < 16; ++jt) {
        v8f acc = {};
        acc = __builtin_amdgcn_wmma_f32_16x16x4_f32(
            /*neg_a=*/false, amat, /*neg_b=*/false, b[jt],
            /*c_mod=*/(short)0, acc, /*reuse_a=*/false, /*reuse_b=*/false);

        if (lane < 16) {
            float* dst = dstBase + 16 * jt;
            #pragma unroll
            for (int r = 0; r < 4; ++r)
                __builtin_nontemporal_store(acc[r], dst + r * PLANE);
        }
    }
}

extern "C" void kernel_launch(void* const* d_in, const int* in_sizes, int n_in,
                              void* d_out, int out_size, void* d_ws, size_t ws_size,
                              hipStream_t stream)
{
    const float* x = (const float*)d_in[0];
    float* out     = (float*)d_out;

    // 96 images * 256 output rows = 24576 waves; 8 waves per 256-thread block.
    const int total_waves = N_IMG * OUT_H;
    const int blocks = total_waves / 8;   // 3072

    haar_dwt2_wmma_kernel<<<blocks, 256, 0, stream>>>(x, out);
}